// MyGCN_v4_55173149885091
// MI455X (gfx1250) — compile-verified
//
#include <hip/hip_runtime.h>
#include <math.h>

typedef float v2f __attribute__((ext_vector_type(2)));
typedef float v8f __attribute__((ext_vector_type(8)));

#define N_NODES 100000
#define N_EDGES 3200000
#define LD 16                      // padded feature stride for all intermediates
#define NTILES ((N_NODES + 15) / 16)   // 6250 exact

// ---------------------------------------------------------------------------
// Generic FP32 WMMA GEMM:  s[node][0..15] = h[node][0..K-1] @ W[K][N]
// W staged in LDS transposed + zero padded to [16][Kpad] (stride Kpad+4).
// One 16x16 output tile per wave via V_WMMA_F32_16X16X4_F32, K-loop step 4.
// ---------------------------------------------------------------------------
__global__ __launch_bounds__(256)
void gemm_wmma_f32(const float* __restrict__ h, const float* __restrict__ W,
                   float* __restrict__ s, int K, int Kpad, int N, int h_ld) {
  extern __shared__ float ldsW[];            // ldsW[n*stride + k], n=0..15
  const int stride = Kpad + 4;               // even -> 8B-aligned float2 reads
  for (int i = threadIdx.x; i < 16 * Kpad; i += blockDim.x) {
    const int k = i % Kpad, n = i / Kpad;
    ldsW[n * stride + k] = (n < N && k < K) ? W[k * N + n] : 0.0f;
  }
  __syncthreads();

  const int lane  = threadIdx.x & 31;
  const int wave  = threadIdx.x >> 5;
  const int nwv   = blockDim.x >> 5;
  const int n     = lane & 15;               // A row / B,C,D column index
  const int hi    = lane >> 4;               // 0: K+{0,1}, 1: K+{2,3}

  for (int tile = blockIdx.x * nwv + wave; tile < NTILES; tile += gridDim.x * nwv) {
    const float* arow = h + (size_t)(tile * 16 + n) * h_ld + 2 * hi;
    const float* bcol = ldsW + n * stride + 2 * hi;
    v8f c = {0.f, 0.f, 0.f, 0.f, 0.f, 0.f, 0.f, 0.f};
    for (int k0 = 0; k0 < Kpad; k0 += 4) {
      v2f a = *(const v2f*)(arow + k0);      // A[M=n][k0+2hi .. +1]
      v2f b = *(const v2f*)(bcol + k0);      // B[k0+2hi .. +1][N=n]
      c = __builtin_amdgcn_wmma_f32_16x16x4_f32(
              /*neg_a=*/false, a, /*neg_b=*/false, b,
              /*c_mod=*/(short)0, c, /*reuse_a=*/false, /*reuse_b=*/false);
    }
    // D: VGPR j -> row tile*16 + j + 8*hi, col n
    float* srow = s + ((size_t)tile * 16 + 8 * hi) * LD + n;
#pragma unroll
    for (int j = 0; j < 8; ++j) srow[j * LD] = c[j];
  }
}

__global__ void zero_f32(float* __restrict__ p, int n) {
  int i = blockIdx.x * blockDim.x + threadIdx.x;
  if (i < n) p[i] = 0.0f;
}

// m = edge_val * s[edge_src]; agg[edge_dst] += m   (segment_sum via f32 atomics,
// accumulator is L2-resident: 100k x 16 floats = 6.4 MB << 192 MB L2)
template <int N>
__global__ void edge_scatter(const float* __restrict__ s, const float* __restrict__ ev,
                             const int* __restrict__ esrc, const int* __restrict__ edst,
                             float* __restrict__ agg) {
  int e = blockIdx.x * blockDim.x + threadIdx.x;
  if (e >= N_EDGES) return;
  const int src = esrc[e];
  const int dst = edst[e];
  const float v = ev[e];
  const float* sr = s + (size_t)src * LD;
  float* ar = agg + (size_t)dst * LD;
#pragma unroll
  for (int f = 0; f < N; ++f) atomicAdd(ar + f, v * sr[f]);
}

// h = act(agg + b); padding columns [N..15] forced to zero so the next layer's
// WMMA A-fragments read zeros for padded K.
// ACT: 0 = identity, 1 = relu, 2 = tanhshrink
template <int N, int ACT>
__global__ void bias_act(const float* __restrict__ agg, const float* __restrict__ b,
                         float* __restrict__ h) {
  int i = blockIdx.x * blockDim.x + threadIdx.x;
  if (i >= N_NODES * LD) return;
  const int col = i & (LD - 1);
  float v = 0.0f;
  if (col < N) {
    v = agg[i] + b[col];
    if (ACT == 1) v = fmaxf(v, 0.0f);
    else if (ACT == 2) v = v - tanhf(v);
  }
  h[i] = v;
}

// Final layer: out[node][0..6] = agg[node][0..6] + b  (packed ld = 7)
__global__ void bias_final7(const float* __restrict__ agg, const float* __restrict__ b,
                            float* __restrict__ out) {
  int i = blockIdx.x * blockDim.x + threadIdx.x;
  if (i >= N_NODES * 7) return;
  const int node = i / 7, col = i % 7;
  out[i] = agg[(size_t)node * LD + col] + b[col];
}

// ---------------------------------------------------------------------------
extern "C" void kernel_launch(void* const* d_in, const int* in_sizes, int n_in,
                              void* d_out, int out_size, void* d_ws, size_t ws_size,
                              hipStream_t stream) {
  (void)in_sizes; (void)n_in; (void)out_size; (void)ws_size;

  const float* x    = (const float*)d_in[0];   // [100000, 512]
  const float* ev   = (const float*)d_in[1];   // [3.2M]
  const int*   esrc = (const int*)d_in[2];
  const int*   edst = (const int*)d_in[3];
  const float* W[6];
  const float* B[6];
  for (int i = 0; i < 6; ++i) { W[i] = (const float*)d_in[4 + 2 * i];
                                B[i] = (const float*)d_in[5 + 2 * i]; }
  float* out = (float*)d_out;

  float* ws   = (float*)d_ws;
  const size_t S = (size_t)N_NODES * LD;       // 1.6M floats per buffer
  float* sbuf = ws;                            // gemm output (pre-scatter)
  float* abuf = ws + S;                        // scatter accumulator
  float* hbuf = ws + 2 * S;                    // activated node features

  const int gemmBlocks = 256, gemmThreads = 256;
  const int zeroGrid   = (int)((S + 255) / 256);
  const int edgeGrid   = (N_EDGES + 255) / 256;
  const int biasGrid   = (N_NODES * LD + 255) / 256;

  const int dimsIn[6]  = {512, 12, 10, 8, 6, 4};
  const int dimsOut[6] = { 12, 10,  8, 6, 4, 7};

  for (int l = 0; l < 6; ++l) {
    const int K    = dimsIn[l];
    const int Kpad = (K + 3) & ~3;
    const int N    = dimsOut[l];
    const int h_ld = (l == 0) ? 512 : LD;
    const float* hin = (l == 0) ? x : hbuf;
    const size_t shmem = (size_t)16 * (Kpad + 4) * sizeof(float);

    gemm_wmma_f32<<<gemmBlocks, gemmThreads, shmem, stream>>>(
        hin, W[l], sbuf, K, Kpad, N, h_ld);
    zero_f32<<<zeroGrid, 256, 0, stream>>>(abuf, (int)S);

    switch (N) {
      case 12: edge_scatter<12><<<edgeGrid, 256, 0, stream>>>(sbuf, ev, esrc, edst, abuf); break;
      case 10: edge_scatter<10><<<edgeGrid, 256, 0, stream>>>(sbuf, ev, esrc, edst, abuf); break;
      case  8: edge_scatter< 8><<<edgeGrid, 256, 0, stream>>>(sbuf, ev, esrc, edst, abuf); break;
      case  6: edge_scatter< 6><<<edgeGrid, 256, 0, stream>>>(sbuf, ev, esrc, edst, abuf); break;
      case  4: edge_scatter< 4><<<edgeGrid, 256, 0, stream>>>(sbuf, ev, esrc, edst, abuf); break;
      case  7: edge_scatter< 7><<<edgeGrid, 256, 0, stream>>>(sbuf, ev, esrc, edst, abuf); break;
    }

    switch (l) {
      case 0: bias_act<12, 0><<<biasGrid, 256, 0, stream>>>(abuf, B[0], hbuf); break;
      case 1: bias_act<10, 1><<<biasGrid, 256, 0, stream>>>(abuf, B[1], hbuf); break;
      case 2: bias_act< 8, 2><<<biasGrid, 256, 0, stream>>>(abuf, B[2], hbuf); break;
      case 3: bias_act< 6, 2><<<biasGrid, 256, 0, stream>>>(abuf, B[3], hbuf); break;
      case 4: bias_act< 4, 0><<<biasGrid, 256, 0, stream>>>(abuf, B[4], hbuf); break;
      case 5: {
        const int g = (N_NODES * 7 + 255) / 256;
        bias_final7<<<g, 256, 0, stream>>>(abuf, B[5], out);
        break;
      }
    }
  }
}